// OrdinalContrastiveLoss_sm_36266703848099
// MI455X (gfx1250) — compile-verified
//
#include <hip/hip_runtime.h>
#include <math.h>

typedef __attribute__((ext_vector_type(2))) float v2f;
typedef __attribute__((ext_vector_type(8))) float v8f;

#define N_ROWS 512
#define K_DIM  256

// ---------------------------------------------------------------------------
// Kernel 1: per-row L2 norms of P [512 x 256]
// ---------------------------------------------------------------------------
__global__ void norms_kernel(const float* __restrict__ P, float* __restrict__ norms) {
    const int row = blockIdx.x;
    const int tid = threadIdx.x;
    __shared__ float red[256];
    const float* p = P + row * K_DIM;
    float s = 0.f;
    for (int k = tid; k < K_DIM; k += 256) {
        float v = p[k];
        s += v * v;
    }
    red[tid] = s;
    __syncthreads();
    for (int off = 128; off > 0; off >>= 1) {
        if (tid < off) red[tid] += red[tid + off];
        __syncthreads();
    }
    if (tid == 0) norms[row] = sqrtf(red[0]);
}

// ---------------------------------------------------------------------------
// Kernel 2: cosine matrix via fp32 WMMA.
// One wave per 16x16 output tile; K accumulated 4 at a time with
// v_wmma_f32_16x16x4_f32.  1024 tiles total -> 128 blocks x 8 waves.
//
// A-fragment (16x4, MxK) lane layout: lanes 0-15 hold M=lane, K=0..1 in
// VGPR0..1; lanes 16-31 hold K=2..3.  B-fragment (4x16, KxN) mirrors it with
// N=lane%16.  For G = P*P^T both fragments read P with the identical per-lane
// pattern:  P[base + lane%16][kk + 2*(lane/16) + {0,1}]  (one b64 load each).
// ---------------------------------------------------------------------------
__global__ void gram_wmma_kernel(const float* __restrict__ P,
                                 const float* __restrict__ norms,
                                 float* __restrict__ cosM) {
    const int lane = threadIdx.x & 31;
    const int wave = threadIdx.x >> 5;
    const int tile = blockIdx.x * 8 + wave;      // 0..1023, exact cover
    const int ti = tile >> 5;                    // row-tile
    const int tj = tile & 31;                    // col-tile
    const int row0 = ti * 16;
    const int col0 = tj * 16;
    const int r = lane & 15;
    const int h = lane >> 4;

    const float* aptr = P + (row0 + r) * K_DIM + 2 * h;
    const float* bptr = P + (col0 + r) * K_DIM + 2 * h;

    v8f acc = {};
#pragma unroll 8
    for (int kk = 0; kk < K_DIM; kk += 4) {
        v2f a = *(const v2f*)(aptr + kk);
        v2f b = *(const v2f*)(bptr + kk);
        // 8 args: (neg_a, A, neg_b, B, c_mod, C, reuse_a, reuse_b)
        acc = __builtin_amdgcn_wmma_f32_16x16x4_f32(
            false, a, false, b, (short)0, acc, false, false);
    }

    // C/D layout: lane groups 0-15 / 16-31 give M = v + 8*h, N = r.
    const float nj = norms[col0 + r];
#pragma unroll
    for (int v = 0; v < 8; ++v) {
        const int i = row0 + v + 8 * h;
        const float denom = fmaxf(norms[i] * nj, 1e-8f);
        cosM[i * N_ROWS + (col0 + r)] = acc[v] / denom;
    }
}

// ---------------------------------------------------------------------------
// Kernel 3a: zero the (poisoned) scalar output before atomic accumulation.
// ---------------------------------------------------------------------------
__global__ void zero_out_kernel(float* __restrict__ out) { out[0] = 0.f; }

// ---------------------------------------------------------------------------
// Kernel 3b: triplet reduction.  One block per anchor i.
// arow[k] = cos[i,k] + d*|t_i - t_k| for negatives k, else -1e30 (relu -> 0,
// matching the reference's "infs relu to 0" masking).  Threads parallelize
// over positive j; 512-deep fmax loop over k runs out of LDS.
// ---------------------------------------------------------------------------
__global__ void loss_kernel(const float* __restrict__ cosM,
                            const int* __restrict__ tgt,
                            const float* __restrict__ rawd,
                            float* __restrict__ out) {
    const int i = blockIdx.x;
    const int tid = threadIdx.x;
    __shared__ float srow[N_ROWS];
    __shared__ float arow[N_ROWS];
    __shared__ int   st[N_ROWS];
    __shared__ float red[256];

    for (int j = tid; j < N_ROWS; j += 256) {
        srow[j] = cosM[i * N_ROWS + j];
        st[j] = tgt[j];
    }
    __syncthreads();

    const int ti_lab = st[i];
    const float raw = rawd[0];
    const float dd = (raw > 20.f) ? raw : log1pf(expf(raw));   // softplus

    for (int k = tid; k < N_ROWS; k += 256) {
        const int tk = st[k];
        arow[k] = (tk != ti_lab)
                      ? srow[k] + dd * fabsf((float)(ti_lab - tk))
                      : -1e30f;
    }
    __syncthreads();

    // counts (cheap; every thread computes redundantly from LDS)
    float negc = 0.f, posc = 0.f;
    for (int k = 0; k < N_ROWS; ++k) {
        const int tk = st[k];
        if (tk != ti_lab)     negc += 1.f;
        else if (k != i)      posc += 1.f;
    }

    float tsum = 0.f;
    for (int j = tid; j < N_ROWS; j += 256) {
        if (st[j] == ti_lab && j != i) {        // positive j
            const float c = srow[j];
            float s = 0.f;
#pragma unroll 8
            for (int k = 0; k < N_ROWS; ++k)
                s += fmaxf(arow[k] - c, 0.f);
            tsum += s;
        }
    }
    red[tid] = tsum;
    __syncthreads();
    for (int off = 128; off > 0; off >>= 1) {
        if (tid < off) red[tid] += red[tid + off];
        __syncthreads();
    }
    if (tid == 0) {
        const float per_i = (red[0] / fmaxf(negc, 1.f)) / fmaxf(negc * posc, 1.f);
        atomicAdd(out, per_i / (float)N_ROWS);
    }
}

// ---------------------------------------------------------------------------
// Launch
// ---------------------------------------------------------------------------
extern "C" void kernel_launch(void* const* d_in, const int* in_sizes, int n_in,
                              void* d_out, int out_size, void* d_ws, size_t ws_size,
                              hipStream_t stream) {
    (void)in_sizes; (void)n_in; (void)out_size; (void)ws_size;

    const float* P    = (const float*)d_in[0];   // prediction [512,256] f32
    const float* rawd = (const float*)d_in[1];   // raw margin scalar  f32
    const int*   tgt  = (const int*)d_in[2];     // target labels [512] i32

    float* out = (float*)d_out;
    float* ws  = (float*)d_ws;
    float* norms = ws;                           // 512 floats
    float* cosM  = ws + N_ROWS;                  // 512*512 floats (1 MB)

    norms_kernel<<<N_ROWS, 256, 0, stream>>>(P, norms);
    gram_wmma_kernel<<<128, 256, 0, stream>>>(P, norms, cosM);
    zero_out_kernel<<<1, 1, 0, stream>>>(out);
    loss_kernel<<<N_ROWS, 256, 0, stream>>>(cosM, tgt, rawd, out);
}